// ChaoticRNN_12678743458140
// MI455X (gfx1250) — compile-verified
//
#include <hip/hip_runtime.h>
#include <hip/hip_bf16.h>
#include <math.h>

// ---------------------------------------------------------------------------
// ChaoticRNN on MI455X (gfx1250), fp32 WMMA path.
//   Phase 1: out[:, t, :] = x @ Wi + B            (HBM-bound big GEMM)
//   Phase 2: ONE persistent kernel, 512 steps:
//              out[:, t, :] = tanh(out[:, t, :] + out[:, t-1, :] @ Wh)
//            - Wh slab (1024x16 per WG = 64KB) LDS-resident for all steps
//            - device-wide barrier via monotonic atomic counter in d_ws
//            - ht_final written in the last step (no copy kernel)
// ---------------------------------------------------------------------------

typedef __attribute__((ext_vector_type(2))) float v2f;
typedef __attribute__((ext_vector_type(8))) float v8f;

static __device__ __forceinline__ v8f wmma4(v2f a, v2f b, v8f c) {
    // V_WMMA_F32_16X16X4_F32 : D = A(16x4) * B(4x16) + C(16x16), all fp32
    return __builtin_amdgcn_wmma_f32_16x16x4_f32(
        /*neg_a=*/false, a, /*neg_b=*/false, b,
        /*c_mod=*/(short)0, c, /*reuse_a=*/false, /*reuse_b=*/false);
}

// ---------------------------------------------------------------------------
// Phase 1: out[M=32768, N=1024] = x[M, K=512] @ Wi[K, N] + bias[N]
// Grid: (N/64, M/16), block 128 (4 waves, one 16x16 tile per wave).
// A-frag: m=lane&15, K-pair k0+2*(lane>=16)  -> contiguous float2 load
// B-frag: n=lane&15, two rows of Wi          -> coalesced across lanes
// ---------------------------------------------------------------------------
__global__ void __launch_bounds__(128)
rnn_xw_gemm(const float* __restrict__ x, const float* __restrict__ Wi,
            const float* __restrict__ bias, float* __restrict__ out) {
    const int K = 512, N = 1024;
    const int wave = threadIdx.x >> 5;
    const int lane = threadIdx.x & 31;
    const int n0 = blockIdx.x * 64 + wave * 16;
    const int m0 = blockIdx.y * 16;
    const int lm = lane & 15;
    const int kh = (lane >> 4) << 1;

    const float* Arow = x + (size_t)(m0 + lm) * K;
    const float* Bcol = Wi + (n0 + lm);

    v8f c0 = {}, c1 = {};
    for (int k0 = 0; k0 < K; k0 += 64) {
        int kp = (k0 + 64 < K) ? (k0 + 64) : k0;
        __builtin_prefetch(Bcol + (size_t)(kp + kh) * N, 0, 1);
#pragma unroll
        for (int k = k0; k < k0 + 64; k += 8) {
            v2f a0 = *(const v2f*)(Arow + k + kh);
            v2f b0;
            b0.x = Bcol[(size_t)(k + kh) * N];
            b0.y = Bcol[(size_t)(k + kh + 1) * N];
            c0 = wmma4(a0, b0, c0);
            v2f a1 = *(const v2f*)(Arow + k + 4 + kh);
            v2f b1;
            b1.x = Bcol[(size_t)(k + 4 + kh) * N];
            b1.y = Bcol[(size_t)(k + 4 + kh + 1) * N];
            c1 = wmma4(a1, b1, c1);
        }
    }
    v8f c = c0 + c1;

    const int n = n0 + lm;
    const int mbase = m0 + ((lane >> 4) << 3);
    const float bv = bias[n];
#pragma unroll
    for (int r = 0; r < 8; ++r) {
        out[(size_t)(mbase + r) * N + n] = c[r] + bv;
    }
}

// ---------------------------------------------------------------------------
// Counter init (replay-safe reset of the device-wide barrier counter).
// ---------------------------------------------------------------------------
__global__ void rnn_init_counter(unsigned* __restrict__ cnt) {
    if (threadIdx.x == 0 && blockIdx.x == 0) *cnt = 0u;
}

// ---------------------------------------------------------------------------
// Phase 2 (persistent): 64 WGs, each owns n-cols [16*bx, 16*bx+16).
// Block = 128 (4 waves); wave w computes the 16x16 tile for batch rows
// [16w, 16w+16) against the WG's 16 columns, K = 1024 per step.
// Wh slab lives in LDS (64KB, loaded once, reused 512x).
// h_{t-1} is read from out[:, t-1, :] (L2-resident, 256KB).
// ---------------------------------------------------------------------------
__global__ void __launch_bounds__(128)
rnn_recurrence(const float* __restrict__ Wh, float* __restrict__ out,
               float* __restrict__ ht_final, unsigned* __restrict__ barrier_cnt) {
    const int K = 1024, N = 1024, T = 512;
    const int NWG = 64;

    __shared__ float whs[K * 16];   // [k][n_local], 64KB, conflict-free reads

    const int tid  = threadIdx.x;
    const int wave = tid >> 5;          // 0..3 -> batch tile
    const int lane = tid & 31;
    const int lm   = lane & 15;         // m for A-frag, n for B-frag
    const int hi   = lane >> 4;         // 0/1
    const int kh   = hi << 1;           // K-pair offset 0/2
    const int n0   = blockIdx.x * 16;   // this WG's column slab

    // One-time Wh slab preload: whs[k*16 + nl] = Wh[k][n0 + nl]
    for (int i = tid; i < K * 16; i += 128) {
        whs[i] = Wh[(size_t)(i >> 4) * N + n0 + (i & 15)];
    }
    __syncthreads();

    const int m_lane = wave * 16 + lm;          // A row for this lane
    const size_t rowstride = (size_t)T * N;     // batch stride in out

    for (int t = 0; t < T; ++t) {
        v8f c0 = {}, c1 = {};
        if (t > 0) {  // uniform branch: EXEC all-ones for WMMA
            const float* Arow = out + (size_t)m_lane * rowstride
                                    + (size_t)(t - 1) * N;   // h_{t-1}
#pragma unroll 8
            for (int k = 0; k < K; k += 8) {
                v2f a0 = *(const v2f*)(Arow + k + kh);
                v2f b0;
                b0.x = whs[(k + kh) * 16 + lm];
                b0.y = whs[(k + kh + 1) * 16 + lm];
                c0 = wmma4(a0, b0, c0);
                v2f a1 = *(const v2f*)(Arow + k + 4 + kh);
                v2f b1;
                b1.x = whs[(k + 4 + kh) * 16 + lm];
                b1.y = whs[(k + 4 + kh + 1) * 16 + lm];
                c1 = wmma4(a1, b1, c1);
            }
        }
        v8f c = c0 + c1;

        // Epilogue: h_t = tanh(xW_t + h_{t-1}@Wh), written in place over xW_t.
        const int n  = n0 + lm;
        const int mb = wave * 16 + hi * 8;
#pragma unroll
        for (int r = 0; r < 8; ++r) {
            size_t idx = (size_t)(mb + r) * rowstride + (size_t)t * N + n;
            float v = tanhf(out[idx] + c[r]);
            out[idx] = v;
            if (t == T - 1) {
                ht_final[(size_t)(mb + r) * N + n] = v;
            }
        }

        // ---- device-wide barrier: all 64 WGs must finish step t ----
        __threadfence();      // make this thread's stores device-visible
        __syncthreads();      // all threads of WG past their fence
        if (tid == 0) {
            __hip_atomic_fetch_add(barrier_cnt, 1u, __ATOMIC_RELEASE,
                                   __HIP_MEMORY_SCOPE_AGENT);
            const unsigned target = (unsigned)NWG * (unsigned)(t + 1);
            while (__hip_atomic_load(barrier_cnt, __ATOMIC_ACQUIRE,
                                     __HIP_MEMORY_SCOPE_AGENT) < target) {
                __builtin_amdgcn_s_sleep(2);
            }
        }
        __syncthreads();      // release whole WG into step t+1
    }
}

extern "C" void kernel_launch(void* const* d_in, const int* in_sizes, int n_in,
                              void* d_out, int out_size, void* d_ws, size_t ws_size,
                              hipStream_t stream) {
    const float* x  = (const float*)d_in[0];  // [64, 512, 512]
    const float* Wi = (const float*)d_in[1];  // [512, 1024]
    const float* Wh = (const float*)d_in[2];  // [1024, 1024]
    const float* B  = (const float*)d_in[3];  // [1024]
    float* out = (float*)d_out;               // [64*512*1024] outs ++ [64*1024] ht

    const int BS = 64, T = 512, H = 1024;
    const int M1 = BS * T;  // 32768

    // Phase 1: big GEMM x@Wi + B  -> out (also serves as xW buffer)
    dim3 g1(H / 64, M1 / 16);
    rnn_xw_gemm<<<g1, 128, 0, stream>>>(x, Wi, B, out);

    // Phase 2: reset barrier counter, then one persistent kernel for all steps
    unsigned* cnt = (unsigned*)d_ws;
    rnn_init_counter<<<1, 32, 0, stream>>>(cnt);

    float* ht_final = out + (size_t)BS * T * H;
    rnn_recurrence<<<H / 16, 128, 0, stream>>>(Wh, out, ht_final, cnt);
}